// compute_miou_41497974014422
// MI455X (gfx1250) — compile-verified
//
#include <hip/hip_runtime.h>

typedef __attribute__((ext_vector_type(4))) int   v4i;
typedef __attribute__((ext_vector_type(2))) float v2f;
typedef __attribute__((ext_vector_type(8))) float v8f;

#define NJ     704      // 22 rows * 32-stride joint histogram (padded)
#define NWAVES 8
#define TPB    256

__device__ __forceinline__ void acc_elem(unsigned* base, int t, int p) {
  atomicAdd(&base[(t << 5) | p], 1u);   // joint bin: one LDS atomic per element
}

template<bool USE_ATOMIC>
__global__ void joint_hist_kernel(const int* __restrict__ ypred,
                                  const int* __restrict__ ytrue,
                                  unsigned* __restrict__ out, // partials (b*NJ+i) or global joint
                                  int n4, int n) {
  __shared__ unsigned sh[NWAVES * NJ];
  const int tid = threadIdx.x;
  for (int i = tid; i < NWAVES * NJ; i += TPB) sh[i] = 0;
  __syncthreads();

  unsigned* base = &sh[(tid >> 5) * NJ];   // wave-private copy (wave32)
  const v4i* yp4 = (const v4i*)ypred;
  const v4i* yt4 = (const v4i*)ytrue;
  const int stride = TPB * gridDim.x;

  for (int i = blockIdx.x * TPB + tid; i < n4; i += stride) {
    v4i p = __builtin_nontemporal_load(&yp4[i]);   // global_load_b128, NT (stream > L2)
    v4i t = __builtin_nontemporal_load(&yt4[i]);
    acc_elem(base, t.x, p.x);
    acc_elem(base, t.y, p.y);
    acc_elem(base, t.z, p.z);
    acc_elem(base, t.w, p.w);
  }
  if (blockIdx.x == 0 && tid == 0) {               // scalar tail (n % 4)
    for (int i = n4 << 2; i < n; ++i) acc_elem(base, ytrue[i], ypred[i]);
  }
  __syncthreads();

  for (int i = tid; i < NJ; i += TPB) {
    unsigned s = 0;
    #pragma unroll
    for (int w = 0; w < NWAVES; ++w) s += sh[w * NJ + i];
    if (USE_ATOMIC) { if (s) atomicAdd(&out[i], s); }
    else            out[(size_t)blockIdx.x * NJ + i] = s;   // deterministic partials
  }
}

__global__ void reduce_partials(const unsigned* __restrict__ parts,
                                unsigned* __restrict__ joint, int nblocks) {
  const int i = blockIdx.x * blockDim.x + threadIdx.x;
  if (i >= NJ) return;
  unsigned s = 0;
  #pragma unroll 8
  for (int b = 0; b < nblocks; ++b) s += parts[(size_t)b * NJ + i];
  joint[i] = s;
}

__global__ void zero_joint(unsigned* g) {
  const int i = blockIdx.x * blockDim.x + threadIdx.x;
  if (i < NJ) g[i] = 0;
}

// One wave. Per-class IoU from the joint histogram, then a single
// V_WMMA_F32_16X16X4_F32 performs BOTH reductions (sum(iou*present), sum(present)):
// A[m][0]/A[m][2] carry x = iou*present, A[m][1]/A[m][3] carry present;
// B selects x into columns 0-7 and present into columns 8-15.
__global__ void finalize_kernel(const unsigned* __restrict__ joint, float* __restrict__ out) {
  const int j = threadIdx.x;          // 0..31, EXEC all ones for WMMA
  float x = 0.f, p = 0.f;
  if (j < 21) {
    const int c = j + 1;
    unsigned ct = 0, cp = 0;
    #pragma unroll
    for (int k = 0; k < 22; ++k) {
      ct += joint[(c << 5) | k];      // row sum  -> cnt_true[c]
      cp += joint[(k << 5) | c];      // col sum  -> cnt_pred[c]
    }
    const unsigned ci = joint[(c << 5) | c];   // diagonal -> intersection
    const float fct = (float)ct, fcp = (float)cp, fci = (float)ci;
    const float uni = fct + fcp - fci;
    const float iou = (fci + 1e-6f) / (uni + 1e-6f);
    p = (fct > 0.f) ? 1.f : 0.f;
    x = iou * p;
    out[1 + j] = iou;
  }
  // A layout (32-bit 16x4): lanes 0-15 -> K=0 (a.x), K=1 (a.y); lanes 16-31 -> K=2, K=3
  v2f a; a.x = x; a.y = p;
  // B layout (4x16): b.x = rows K=0/K=2, b.y = rows K=1/K=3; column n = lane & 15
  const int n = j & 15;
  v2f b; b.x = (n < 8) ? 1.f : 0.f;
         b.y = (n < 8) ? 0.f : 1.f;
  v8f c = {};
  v8f d = __builtin_amdgcn_wmma_f32_16x16x4_f32(
      /*neg_a=*/false, a, /*neg_b=*/false, b,
      /*c_mod=*/(short)0, c, /*reuse_a=*/false, /*reuse_b=*/false);
  float s = d[0] + d[1] + d[2] + d[3] + d[4] + d[5] + d[6] + d[7];
  // D[m][n]: lanes 0-15 hold M=0..7 (summed above), lanes 16-31 hold M=8..15
  const float S1 = __shfl(s, 0, 32) + __shfl(s, 16, 32);   // col 0: sum(iou*present)
  const float S2 = __shfl(s, 8, 32) + __shfl(s, 24, 32);   // col 8: sum(present)
  if (j == 0) out[0] = S1 / fmaxf(S2, 1.0f);
}

extern "C" void kernel_launch(void* const* d_in, const int* in_sizes, int n_in,
                              void* d_out, int out_size, void* d_ws, size_t ws_size,
                              hipStream_t stream) {
  const int* ypred = (const int*)d_in[0];
  const int* ytrue = (const int*)d_in[1];
  float* out = (float*)d_out;
  unsigned* ws = (unsigned*)d_ws;
  const int n  = in_sizes[0];
  const int n4 = n >> 2;

  const size_t binbytes = (size_t)NJ * sizeof(unsigned);
  int B = 1024;                                   // hist blocks (grid-stride)
  while (B > 32 && (size_t)(B + 1) * binbytes > ws_size) B >>= 1;

  if ((size_t)(B + 1) * binbytes <= ws_size) {
    // Deterministic two-phase: per-block partials (all overwritten -> no init), then reduce.
    unsigned* joint = ws;
    unsigned* parts = ws + NJ;
    joint_hist_kernel<false><<<B, TPB, 0, stream>>>(ypred, ytrue, parts, n4, n);
    reduce_partials<<<(NJ + 63) / 64, 64, 0, stream>>>(parts, joint, B);
    finalize_kernel<<<1, 32, 0, stream>>>(joint, out);
  } else {
    // Tiny-ws fallback: integer global atomics (order-independent -> still deterministic).
    unsigned* joint = ws;
    zero_joint<<<(NJ + 127) / 128, 128, 0, stream>>>(joint);
    joint_hist_kernel<true><<<1024, TPB, 0, stream>>>(ypred, ytrue, joint, n4, n);
    finalize_kernel<<<1, 32, 0, stream>>>(joint, out);
  }
}